// TripletOHNM_63574105916124
// MI455X (gfx1250) — compile-verified
//
#include <hip/hip_runtime.h>

typedef __attribute__((ext_vector_type(2))) float v2f;
typedef __attribute__((ext_vector_type(8))) float v8f;

#define BB 256
#define NN 131072
#define DD 128
#define SLAB 64
#define NSLAB (NN / SLAB)        // 2048
#define KSTEPS (DD / 4)          // 32 wmma k-steps
#define TOPK 5
#define MARGIN 0.3f
#define NEG_FILL (-100.0f)
#define YT_STRIDE (DD + 4)       // pad: conflict-free v2f fragment reads
#define SIMS_STRIDE (SLAB + 2)   // pad: conflict-free tile stores + scans
#define NEG_INF (-1.0e30f)

// Phase 1: one block per 64-column slab of y; computes sim for all 256 rows
// against the slab with f32 WMMA, masks the positive, extracts per-row top-5.
__global__ __launch_bounds__(256) void triplet_phase1(
    const float* __restrict__ x, const float* __restrict__ y,
    const int* __restrict__ pos, float* __restrict__ simP,
    float* __restrict__ cand)
{
  __shared__ float yt[SLAB * YT_STRIDE];     // ~33.8 KB
  __shared__ float sims[BB * SIMS_STRIDE];   // ~67.6 KB

  const int tid = threadIdx.x;
  const int lane = tid & 31;
  const int wave = tid >> 5;
  const int slab = blockIdx.x;
  const int colBase = slab * SLAB;

  // Stage y slab into LDS, coalesced float4 (each y row = 512B contiguous).
  for (int e = tid; e < SLAB * (DD / 4); e += 256) {
    const int c = e >> 5;                    // 32 float4 per row
    const int q = e & 31;
    const float4 v = ((const float4*)(y + (size_t)(colBase + c) * DD))[q];
    *(float4*)(&yt[c * YT_STRIDE + 4 * q]) = v;
  }
  __syncthreads();

  const int halfk = lane >> 4;               // 0 or 1
  const int ln16 = lane & 15;

  // Each wave owns 2 row-tiles (32 rows); A-fragments register-cached across
  // the 4 column tiles.
  for (int rti = 0; rti < 2; ++rti) {
    const int rt = wave * 2 + rti;
    const int arow = rt * 16 + ln16;
    const float* xrow = x + (size_t)arow * DD + 2 * halfk;
    v2f afrag[KSTEPS];
#pragma unroll
    for (int kk = 0; kk < KSTEPS; ++kk)
      afrag[kk] = *(const v2f*)(xrow + 4 * kk);

    for (int ct = 0; ct < SLAB / 16; ++ct) {
      const float* yrow = &yt[(ct * 16 + ln16) * YT_STRIDE + 2 * halfk];
      v8f acc = {0.f, 0.f, 0.f, 0.f, 0.f, 0.f, 0.f, 0.f};
#pragma unroll
      for (int kk = 0; kk < KSTEPS; ++kk) {
        const v2f bfrag = *(const v2f*)(yrow + 4 * kk);
        acc = __builtin_amdgcn_wmma_f32_16x16x4_f32(
            false, afrag[kk], false, bfrag, (short)0, acc, false, false);
      }
      // D layout: vgpr j, lanes0-15 -> M=j, lanes16-31 -> M=j+8, N=lane%16
      const int col = colBase + ct * 16 + ln16;
#pragma unroll
      for (int j = 0; j < 8; ++j) {
        const int row = rt * 16 + j + 8 * halfk;
        float v = acc[j];
        if (col == pos[row]) { simP[row] = v; v = NEG_FILL; }  // one lane grid-wide
        sims[row * SIMS_STRIDE + ct * 16 + ln16] = v;
      }
    }
  }
  __syncthreads();

  // Per-row top-5 over the 64-wide slab: each wave handles 32 rows.
  for (int ri = 0; ri < BB / 8; ++ri) {
    const int row = wave * 32 + ri;
    float a0 = sims[row * SIMS_STRIDE + lane];
    float a1 = sims[row * SIMS_STRIDE + 32 + lane];
    if (a1 > a0) { const float t = a0; a0 = a1; a1 = t; }
    float out5[TOPK];
#pragma unroll
    for (int k = 0; k < TOPK; ++k) {
      float m = a0;
#pragma unroll
      for (int s = 16; s > 0; s >>= 1) m = fmaxf(m, __shfl_xor(m, s));
      const unsigned long long bal = __ballot(a0 == m);
      const int leader = __ffsll(bal) - 1;
      if (lane == leader) { a0 = a1; a1 = NEG_INF; }
      out5[k] = m;
    }
    if (lane == 0) {
      float* c = cand + ((size_t)row * NSLAB + slab) * TOPK;
#pragma unroll
      for (int k = 0; k < TOPK; ++k) c[k] = out5[k];
    }
  }
}

// Phase 2: single block merges 2048 slab-candidates per row into global top-5,
// applies hinge + masked softmax, reduces to the scalar mean.
__global__ __launch_bounds__(256) void triplet_phase2(
    const float* __restrict__ simP, const float* __restrict__ cand,
    float* __restrict__ out)
{
  __shared__ float wsum[8];
  const int tid = threadIdx.x;
  const int lane = tid & 31;
  const int wave = tid >> 5;
  float accum = 0.0f;

  for (int ri = 0; ri < 32; ++ri) {
    const int row = wave * 32 + ri;
    const float* cr = cand + (size_t)row * NSLAB * TOPK;
    float t0 = NEG_INF, t1 = NEG_INF, t2 = NEG_INF, t3 = NEG_INF, t4 = NEG_INF;
    for (int e = lane; e < NSLAB * TOPK; e += 32) {
      const float v = cr[e];
      if (v > t4) {
        if (v > t0)      { t4 = t3; t3 = t2; t2 = t1; t1 = t0; t0 = v; }
        else if (v > t1) { t4 = t3; t3 = t2; t2 = t1; t1 = v; }
        else if (v > t2) { t4 = t3; t3 = t2; t2 = v; }
        else if (v > t3) { t4 = t3; t3 = v; }
        else             { t4 = v; }
      }
    }
    float res[TOPK];
#pragma unroll
    for (int k = 0; k < TOPK; ++k) {
      float m = t0;
#pragma unroll
      for (int s = 16; s > 0; s >>= 1) m = fmaxf(m, __shfl_xor(m, s));
      const unsigned long long bal = __ballot(t0 == m);
      const int leader = __ffsll(bal) - 1;
      if (lane == leader) { t0 = t1; t1 = t2; t2 = t3; t3 = t4; t4 = NEG_INF; }
      res[k] = m;
    }
    if (lane == 0) {
      const float sp = simP[row];
      float z[TOPK], loss[TOPK], zmax = NEG_INF;
#pragma unroll
      for (int k = 0; k < TOPK; ++k) {
        loss[k] = fmaxf(0.0f, res[k] - sp + MARGIN);
        z[k] = (loss[k] != 0.0f) ? res[k] : 0.0f;
        zmax = fmaxf(zmax, z[k]);
      }
      float s = 0.0f, num = 0.0f;
#pragma unroll
      for (int k = 0; k < TOPK; ++k) {
        const float p = expf(z[k] - zmax);
        s += p;
        num += loss[k] * p;
      }
      accum += num / s;
    }
  }
  if (lane == 0) wsum[wave] = accum;
  __syncthreads();
  if (tid == 0) {
    float t = 0.0f;
    for (int w = 0; w < 8; ++w) t += wsum[w];
    out[0] = t / (float)(BB * TOPK);
  }
}

extern "C" void kernel_launch(void* const* d_in, const int* in_sizes, int n_in,
                              void* d_out, int out_size, void* d_ws, size_t ws_size,
                              hipStream_t stream) {
  const float* x = (const float*)d_in[0];     // xembs [256,128]
  const float* y = (const float*)d_in[1];     // yembs [131072,128]
  // d_in[2] = target: unused (single positive per row encoded by pos_inds)
  const int* pos = (const int*)d_in[3];       // pos_inds [256]
  // d_in[4] = num_neg (=5, hardcoded)

  float* simP = (float*)d_ws;                 // 256 floats
  float* cand = simP + BB;                    // 256*2048*5 floats (~10.5 MB)

  triplet_phase1<<<NSLAB, 256, 0, stream>>>(x, y, pos, simP, cand);
  triplet_phase2<<<1, 256, 0, stream>>>(simP, cand, (float*)d_out);
}